// PCQActivation_86801289052782
// MI455X (gfx1250) — compile-verified
//
#include <hip/hip_runtime.h>
#include <cstdint>
#include <cstddef>

// ---- problem constants (match reference) ----
#define KC      10            // num clusters
#define NB      128           // batch
#define IMG_F4  100352        // C*H*W/4 = 128*56*56/4 float4 per image
#define BPB     49            // blocks per image
#define TPB     256           // threads per block (8 wave32)
#define ITERS   8             // 49*256*8 == 100352
#define QMAX    255.0f

typedef float f32x4 __attribute__((ext_vector_type(4)));

// ---------- helpers ----------
__device__ __forceinline__ float hswish(float x) {
    return x * fminf(fmaxf(x + 3.0f, 0.0f), 6.0f) * (1.0f / 6.0f);
}

// float atomic min/max via signed/unsigned int ordering trick (works for all
// sign combinations given +inf / -inf initialization).
__device__ __forceinline__ void atomicMinF(float* a, float v) {
    if (v >= 0.0f) atomicMin((int*)a, __float_as_int(v));
    else           atomicMax((unsigned int*)a, __float_as_uint(v));
}
__device__ __forceinline__ void atomicMaxF(float* a, float v) {
    if (v >= 0.0f) atomicMax((int*)a, __float_as_int(v));
    else           atomicMin((unsigned int*)a, __float_as_uint(v));
}

// ---------- kernel 1: init cluster ranges ----------
__global__ void pcq_init(float* __restrict__ ws) {
    int k = threadIdx.x;
    if (k < KC) {
        ws[k]      =  __builtin_inff();   // c_min
        ws[KC + k] = -__builtin_inff();   // c_max
    }
}

// ---------- kernel 2: per-cluster min/max of hardswish(x) ----------
// blockIdx.y = image b (cluster is block-uniform), blockIdx.x = chunk.
// Streams x through LDS with CDNA5 async DMA (double-buffered, wave-private
// 512B tiles), so VMEM latency is hidden by the async engine (ASYNCcnt)
// instead of VGPR-resident load queues.
__global__ __launch_bounds__(TPB) void pcq_minmax(const f32x4* __restrict__ x4,
                                                  const int* __restrict__ seg,
                                                  float* __restrict__ ws) {
    extern __shared__ f32x4 smem[];   // [2][8 waves][32 lanes] tiles + 16f scratch
    const int tid  = threadIdx.x;
    const int w    = tid >> 5;
    const int lane = tid & 31;
    const int b    = blockIdx.y;

    const f32x4* base = x4 + (size_t)b * IMG_F4 + (size_t)blockIdx.x * TPB + tid;

    // prologue: issue tile 0 (each lane copies 16B global -> its LDS slot)
    {
        unsigned lds0 = (unsigned)((w * 32 + lane) * 16);
        uint64_t g = (uint64_t)base;
        asm volatile("global_load_async_to_lds_b128 %0, %1, off"
                     :: "v"(lds0), "v"(g) : "memory");
    }

    float mn =  __builtin_inff();
    float mx = -__builtin_inff();

    for (int i = 0; i < ITERS; ++i) {
        const int buf = i & 1;
        if (i + 1 < ITERS) {
            unsigned ldsn = (unsigned)(((((i + 1) & 1) * 256) + w * 32 + lane) * 16);
            uint64_t g = (uint64_t)(base + (size_t)(i + 1) * (BPB * TPB));
            asm volatile("global_load_async_to_lds_b128 %0, %1, off"
                         :: "v"(ldsn), "v"(g) : "memory");
            asm volatile("s_wait_asynccnt 0x1" ::: "memory");  // tile i done (in-order)
        } else {
            asm volatile("s_wait_asynccnt 0x0" ::: "memory");
        }
        f32x4 v = smem[buf * 256 + w * 32 + lane];
        float a0 = hswish(v.x), a1 = hswish(v.y);
        float a2 = hswish(v.z), a3 = hswish(v.w);
        mn = fminf(mn, fminf(fminf(a0, a1), fminf(a2, a3)));
        mx = fmaxf(mx, fmaxf(fmaxf(a0, a1), fmaxf(a2, a3)));
    }

    // wave32 butterfly reduction
    for (int o = 16; o > 0; o >>= 1) {
        mn = fminf(mn, __shfl_xor(mn, o, 32));
        mx = fmaxf(mx, __shfl_xor(mx, o, 32));
    }

    float* red = (float*)(smem + 512);   // scratch after the 8KB of tiles
    if (lane == 0) { red[w] = mn; red[8 + w] = mx; }
    __syncthreads();

    if (tid == 0) {
        for (int j = 1; j < 8; ++j) {
            mn = fminf(mn, red[j]);
            mx = fmaxf(mx, red[8 + j]);
        }
        const int c = seg[b];
        atomicMinF(&ws[c],      mn);
        atomicMaxF(&ws[KC + c], mx);
    }
}

// ---------- kernel 3: EMA + qparams ----------
__global__ void pcq_qparams(const float* __restrict__ act_range,
                            float* __restrict__ ws,
                            float* __restrict__ out_ar) {
    int k = threadIdx.x;
    if (k < KC) {
        const float SM  = 0.999f;
        const float OMS = (float)(1.0 - 0.999);   // matches python double->f32
        float nmin = act_range[2 * k]     * SM + ws[k]      * OMS;
        float nmax = act_range[2 * k + 1] * SM + ws[KC + k] * OMS;
        float s = (nmax - nmin) / QMAX;
        float z = -rintf(nmin / s);               // round-half-even like jnp.round
        ws[2 * KC + k] = s;
        ws[3 * KC + k] = z;
        out_ar[2 * k]     = nmin;                 // new_act_range[k,0]
        out_ar[2 * k + 1] = nmax;                 // new_act_range[k,1]
    }
}

// ---------- kernel 4: fake-quantize ----------
// Re-reads x (hot in 192MB L2 from pass 2) with NT loads (last-use) and writes
// q with NT b128 stores so the outgoing stream does not evict x from L2.
__global__ __launch_bounds__(TPB) void pcq_quant(const f32x4* __restrict__ x4,
                                                 const int* __restrict__ seg,
                                                 const float* __restrict__ ws,
                                                 f32x4* __restrict__ out) {
    const int b = blockIdx.y;
    const int c = seg[b];                          // block-uniform -> scalar
    const float s = ws[2 * KC + c];
    const float z = ws[3 * KC + c];

    size_t off = (size_t)b * IMG_F4 + (size_t)blockIdx.x * TPB + threadIdx.x;

    for (int i = 0; i < ITERS; ++i) {
        f32x4 v = __builtin_nontemporal_load(&x4[off]);
        f32x4 r;
        {
            float a = hswish(v.x);
            float t = fminf(fmaxf(rintf(a / s + z), 0.0f), QMAX);
            r.x = (t - z) * s;
        }
        {
            float a = hswish(v.y);
            float t = fminf(fmaxf(rintf(a / s + z), 0.0f), QMAX);
            r.y = (t - z) * s;
        }
        {
            float a = hswish(v.z);
            float t = fminf(fmaxf(rintf(a / s + z), 0.0f), QMAX);
            r.z = (t - z) * s;
        }
        {
            float a = hswish(v.w);
            float t = fminf(fmaxf(rintf(a / s + z), 0.0f), QMAX);
            r.w = (t - z) * s;
        }
        __builtin_nontemporal_store(r, &out[off]);
        off += (size_t)(BPB * TPB);
    }
}

// ---------- host launcher ----------
extern "C" void kernel_launch(void* const* d_in, const int* in_sizes, int n_in,
                              void* d_out, int out_size, void* d_ws, size_t ws_size,
                              hipStream_t stream) {
    const float* x         = (const float*)d_in[0];   // [B,C,H,W] f32
    const float* act_range = (const float*)d_in[1];   // [K,2] f32
    const int*   seg       = (const int*)d_in[2];     // [B] i32
    float* out = (float*)d_out;
    float* ws  = (float*)d_ws;                        // [0,K) cmin  [K,2K) cmax
                                                      // [2K,3K) s   [3K,4K) z
    const size_t QELEMS = (size_t)NB * IMG_F4 * 4;    // 51,380,224

    pcq_init<<<1, 32, 0, stream>>>(ws);

    dim3 grid(BPB, NB);
    // dyn LDS: 2 bufs * 8 waves * 512B tiles = 8192B + 64B reduce scratch
    pcq_minmax<<<grid, TPB, 8192 + 64, stream>>>((const f32x4*)x, seg, ws);

    pcq_qparams<<<1, 32, 0, stream>>>(act_range, ws, out + QELEMS);

    pcq_quant<<<grid, TPB, 0, stream>>>((const f32x4*)x, seg, ws, (f32x4*)out);
}